// FeatureDropoutAugmentation_15917148799756
// MI455X (gfx1250) — compile-verified
//
#include <hip/hip_runtime.h>
#include <stdint.h>

typedef float v4f __attribute__((ext_vector_type(4)));

#define AUG_PROB  0.5f
#define DROP_PROB 0.15f

// ---------------------------------------------------------------------------
// Phase A: per-row drop decision. One block per row; rank-by-counting in LDS.
// keep[b*F+f] = 0.0f if feature dropped else 1.0f
// ---------------------------------------------------------------------------
__global__ void drop_mask_kernel(const int* __restrict__ amask,
                                 const float* __restrict__ aug_rand,
                                 const float* __restrict__ perm_scores,
                                 float* __restrict__ keep, int F) {
    extern __shared__ float s_sc[];     // F floats
    __shared__ int s_cnt;
    const int b = blockIdx.x;
    if (threadIdx.x == 0) s_cnt = 0;
    __syncthreads();

    int local_avail = 0;
    for (int f = threadIdx.x; f < F; f += blockDim.x) {
        const int av = (amask[(long)b * F + f] != 0);
        s_sc[f] = av ? perm_scores[(long)b * F + f] : 2.0f;
        local_avail += av;
    }
    atomicAdd(&s_cnt, local_avail);
    __syncthreads();

    const int n_avail = s_cnt;
    int n_drop = (int)((float)n_avail * DROP_PROB);   // trunc, matches int() cast
    n_drop = min(n_drop, n_avail - 1);
    const bool row_active = (aug_rand[b] < AUG_PROB) && (n_avail > 1) && (n_drop > 0);

    for (int f = threadIdx.x; f < F; f += blockDim.x) {
        const float sc = s_sc[f];
        int rank = 0;
        for (int g = 0; g < F; ++g) {                 // LDS broadcast reads
            const float sg = s_sc[g];
            rank += (sg < sc) || ((sg == sc) && (g < f));  // stable tie-break
        }
        const bool av   = (amask[(long)b * F + f] != 0);
        const bool drop = row_active && av && (rank < n_drop);
        keep[(long)b * F + f] = drop ? 0.0f : 1.0f;
    }
}

// ---------------------------------------------------------------------------
// Phase B: streaming apply. Double-buffered async global->LDS staging
// (global_load_async_to_lds_b128, ASYNCcnt), VALU scale, non-temporal store.
// Each lane stages & consumes its own 16B LDS slot -> only asynccnt ordering.
// All staging guards are wave-uniform (EXEC full when the async op issues).
// ---------------------------------------------------------------------------
__global__ void apply_kernel(const v4f* __restrict__ in4,
                             const float* __restrict__ keep,
                             v4f* __restrict__ out4,
                             long nTiles, int tilesPerBlock, int d4, int shift) {
    __shared__ v4f buf[2][256];
    const int tid = threadIdx.x;

    long tile0 = (long)blockIdx.x * tilesPerBlock;
    if (tile0 >= nTiles) return;
    long tend = tile0 + tilesPerBlock;
    if (tend > nTiles) tend = nTiles;

    const unsigned lds0 = (unsigned)(size_t)&buf[0][tid];
    const unsigned lds1 = (unsigned)(size_t)&buf[1][tid];

    // prime the pipeline: stage tile0 into buffer 0
    {
        unsigned long long g =
            (unsigned long long)(uintptr_t)(in4 + (tile0 << 8) + tid);
        asm volatile("global_load_async_to_lds_b128 %0, %1, off th:TH_LOAD_NT"
                     :: "v"(lds0), "v"(g) : "memory");
    }

    int p = 0;
    for (long tc = tile0; tc < tend; ++tc) {
        const bool more = (tc + 1) < tend;            // uniform across block
        if (more) {
            unsigned long long g =
                (unsigned long long)(uintptr_t)(in4 + ((tc + 1) << 8) + tid);
            const unsigned ldsn = (p == 0) ? lds1 : lds0;
            asm volatile("global_load_async_to_lds_b128 %0, %1, off th:TH_LOAD_NT"
                         :: "v"(ldsn), "v"(g) : "memory");
            asm volatile("s_wait_asynccnt 0x1" ::: "memory");
        } else {
            asm volatile("s_wait_asynccnt 0x0" ::: "memory");
        }

        const long idx  = (tc << 8) + tid;            // float4 index
        const long fidx = (shift >= 0) ? (idx >> shift) : (idx / (long)d4);
        const float m   = keep[fidx];

        v4f v = buf[p][tid];
        __builtin_nontemporal_store(v * m, &out4[idx]);
        p ^= 1;
    }
}

// ---------------------------------------------------------------------------
// Tail: scalar fallback for elements not covered by the vectorized kernel
// (not launched for the reference shapes; here for generality).
// ---------------------------------------------------------------------------
__global__ void tail_kernel(const float* __restrict__ in,
                            const float* __restrict__ keep,
                            float* __restrict__ out,
                            long start, long total, int D) {
    long i = start + (long)blockIdx.x * blockDim.x + threadIdx.x;
    if (i < total) {
        long f = i / (long)D;
        out[i] = in[i] * keep[f];
    }
}

extern "C" void kernel_launch(void* const* d_in, const int* in_sizes, int n_in,
                              void* d_out, int out_size, void* d_ws, size_t ws_size,
                              hipStream_t stream) {
    const float* in_f  = (const float*)d_in[0];   // [B, F, D] f32
    const int*   amask = (const int*)d_in[1];     // [B, F]   i32
    const float* aug   = (const float*)d_in[2];   // [B]      f32
    const float* ps    = (const float*)d_in[3];   // [B, F]   f32
    float*       out   = (float*)d_out;
    float*       keep  = (float*)d_ws;            // [B, F] keep multipliers

    const int  B  = in_sizes[2];
    const long BF = (long)in_sizes[1];
    const int  F  = (int)(BF / B);
    const long N  = (long)out_size;               // B*F*D
    const int  D  = (int)(N / BF);

    // Phase A: one block per row
    drop_mask_kernel<<<dim3((unsigned)B), dim3(256), F * sizeof(float), stream>>>(
        amask, aug, ps, keep, F);

    // Phase B: vectorized path requires feature-aligned float4s (D % 4 == 0)
    long nTiles = 0;
    long covered = 0;
    if ((D % 4) == 0) {
        const long N4 = N >> 2;
        nTiles = N4 >> 8;                         // 256 float4 per tile
        const int d4 = D >> 2;
        const int shift = ((d4 & (d4 - 1)) == 0) ? __builtin_ctz((unsigned)d4) : -1;
        if (nTiles > 0) {
            const int TPB = 16;                   // tiles per block (64 KB stream)
            const long grid = (nTiles + TPB - 1) / TPB;
            apply_kernel<<<dim3((unsigned)grid), dim3(256), 0, stream>>>(
                (const v4f*)in_f, keep, (v4f*)out, nTiles, TPB, d4, shift);
        }
        covered = nTiles << 10;                   // floats covered (tiles * 1024)
    }

    if (covered < N) {
        const long rem = N - covered;
        const long grid = (rem + 255) / 256;
        tail_kernel<<<dim3((unsigned)grid), dim3(256), 0, stream>>>(
            in_f, keep, out, covered, N, D);
    }
}